// DRNN_47193100648718
// MI455X (gfx1250) — compile-verified
//
#include <hip/hip_runtime.h>
#include <math.h>

typedef __attribute__((ext_vector_type(2))) float v2f;
typedef __attribute__((ext_vector_type(8))) float v8f;

#define BATCH 512
#define SEQT  512
#define IND   64
#define H     100
#define NT    7        // N tiles of 16 covering padded hidden (112)
#define KB1   16       // K-blocks (of 4) for x contraction: 64/4
#define KBH   28       // K-blocks (of 4) for hidden contraction: 112/4
#define HSTR  116      // LDS row stride for h tiles (conflict-free)
#define XSTR  68       // LDS row stride for x tile (conflict-free)
#define HBUF  (16*HSTR) // one h buffer: 1856 floats

// float offsets into dynamic LDS
#define OFF_W1X 0
#define OFF_W1H (OFF_W1X + NT*KB1*32*2)     // 7168
#define OFF_W2X (OFF_W1H + NT*KBH*32*2)     // 19712
#define OFF_W2H (OFF_W2X + NT*KBH*32*2)     // 32256
#define OFF_H1  (OFF_W2H + NT*KBH*32*2)     // 44800  (2 buffers)
#define OFF_H2  (OFF_H1 + 2*HBUF)           // 48512  (2 buffers)
#define OFF_X   (OFF_H2 + 2*HBUF)           // 52224  (2 buffers)
#define LDS_FLOATS (OFF_X + 2*16*XSTR)      // 54400 floats = 217600 bytes

// Hardware V_TANH_F32 (gfx1250 TRANS op) if the toolchain exposes it.
#if __has_builtin(__builtin_amdgcn_tanhf)
#define TANHF(v) __builtin_amdgcn_tanhf(v)
#elif __has_builtin(__builtin_amdgcn_tanh_f32)
#define TANHF(v) __builtin_amdgcn_tanh_f32(v)
#else
#define TANHF(v) tanhf(v)
#endif

// Store weights into LDS pre-swizzled into the per-lane WMMA B-fragment layout:
// frag index = ((ntile*KB + kblk)*32 + lane), each holding float2 {W[K][N], W[K+1][N]}
// with K = 4*kblk + 2*(lane>>4), N = ntile*16 + (lane&15). Out-of-range -> 0.
__device__ inline void fill_frags(float* dst, const float* W, int Ka, int Na,
                                  int kb, int tid, int nthr) {
    int total = NT * kb * 32;
    for (int i = tid; i < total; i += nthr) {
        int laneI = i & 31;
        int kblk  = (i >> 5) % kb;
        int nt    = (i >> 5) / kb;
        int N = nt * 16 + (laneI & 15);
        int K = kblk * 4 + ((laneI >> 4) << 1);
        float a = 0.f, b = 0.f;
        if (N < Na) {
            if (K     < Ka) a = W[(size_t)K * Na + N];
            if (K + 1 < Ka) b = W[(size_t)(K + 1) * Na + N];
        }
        dst[2 * i + 0] = a;
        dst[2 * i + 1] = b;
    }
}

__device__ inline v2f ld2(const float* p) { return *(const v2f*)p; }

__global__ void __launch_bounds__(224)
drnn_fused_kernel(const float* __restrict__ x,
                  const float* __restrict__ W1x, const float* __restrict__ W1h,
                  const float* __restrict__ b1,
                  const float* __restrict__ W2x, const float* __restrict__ W2h,
                  const float* __restrict__ b2,
                  const float* __restrict__ Wo,  const float* __restrict__ bo,
                  float* __restrict__ outp, float* __restrict__ h1out,
                  float* __restrict__ h2out) {
    extern __shared__ float lds[];
    const int tid  = threadIdx.x;
    const int nthr = blockDim.x;          // 224 = 7 waves
    const int b0   = blockIdx.x * 16;     // batch tile base

    // ---- one-time init: weights -> LDS frag layout, h buffers -> 0 ----
    fill_frags(lds + OFF_W1X, W1x, IND, H, KB1, tid, nthr);
    fill_frags(lds + OFF_W1H, W1h, H,   H, KBH, tid, nthr);
    fill_frags(lds + OFF_W2X, W2x, H,   H, KBH, tid, nthr);
    fill_frags(lds + OFF_W2H, W2h, H,   H, KBH, tid, nthr);
    for (int i = tid; i < 4 * HBUF; i += nthr) lds[OFF_H1 + i] = 0.f;

    // stage x_0 into buffer 0 (16 rows x 64 floats, padded stride 68)
    {
        #pragma unroll
        for (int ii = 0; ii < 2; ++ii) {
            int i = tid + ii * 224;
            if (i < 256) {
                int row = i >> 4, c = (i & 15) << 2;
                float4 v = *(const float4*)(x + ((size_t)(b0 + row) * SEQT + 0) * IND + c);
                *(float4*)(lds + OFF_X + row * XSTR + c) = v;
            }
        }
    }
    __syncthreads();

    const int lane = tid & 31;
    const int wv   = tid >> 5;            // wave id == N-tile id, 0..6
    const int mi   = lane & 15;           // M (or N) index within tile
    const int kh   = (lane >> 4) << 1;    // K sub-offset: 0 or 2
    const int crow = (lane >> 4) << 3;    // C-tile row base: 0 or 8

    float bias1 = 0.f, bias2 = 0.f;
    {
        int n = wv * 16 + mi;
        if (n < H) { bias1 = b1[n]; bias2 = b2[n]; }
    }

    for (int t = 0; t < SEQT; ++t) {
        const int cb = t & 1;          // current h / current x buffer
        const int pb = cb ^ 1;         // previous h / next x buffer

        // --- prefetch x_{t+1} into registers (latency hidden under WMMAs) ---
        float4 xr0 = {}, xr1 = {};
        const bool has1 = (tid < 32);
        if (t + 1 < SEQT) {
            int row = tid >> 4, c = (tid & 15) << 2;
            xr0 = *(const float4*)(x + ((size_t)(b0 + row) * SEQT + (t + 1)) * IND + c);
            if (has1) {
                int j = tid + 224;
                int r2 = j >> 4, c2 = (j & 15) << 2;
                xr1 = *(const float4*)(x + ((size_t)(b0 + r2) * SEQT + (t + 1)) * IND + c2);
            }
        }

        // --- layer 1: acc = x_t @ W1x + h1[prev] @ W1h + b1 ---
        v8f c0, c1;
        #pragma unroll
        for (int r = 0; r < 8; ++r) { c0[r] = bias1; c1[r] = 0.f; }
        {
            const float* xl = lds + OFF_X + cb * (16 * XSTR) + mi * XSTR + kh;
            const float* wp = lds + OFF_W1X + wv * (KB1 * 64) + lane * 2;
            #pragma unroll
            for (int k = 0; k < KB1; k += 2) {
                c0 = __builtin_amdgcn_wmma_f32_16x16x4_f32(
                        false, ld2(xl + 4 * k),       false, ld2(wp + 64 * k),
                        (short)0, c0, false, false);
                c1 = __builtin_amdgcn_wmma_f32_16x16x4_f32(
                        false, ld2(xl + 4 * (k + 1)), false, ld2(wp + 64 * (k + 1)),
                        (short)0, c1, false, false);
            }
            const float* hl = lds + OFF_H1 + pb * HBUF + mi * HSTR + kh;
            wp = lds + OFF_W1H + wv * (KBH * 64) + lane * 2;
            #pragma unroll
            for (int k = 0; k < KBH; k += 2) {
                c0 = __builtin_amdgcn_wmma_f32_16x16x4_f32(
                        false, ld2(hl + 4 * k),       false, ld2(wp + 64 * k),
                        (short)0, c0, false, false);
                c1 = __builtin_amdgcn_wmma_f32_16x16x4_f32(
                        false, ld2(hl + 4 * (k + 1)), false, ld2(wp + 64 * (k + 1)),
                        (short)0, c1, false, false);
            }
        }

        // commit staged x_{t+1} to the other LDS x buffer
        if (t + 1 < SEQT) {
            int row = tid >> 4, c = (tid & 15) << 2;
            *(float4*)(lds + OFF_X + pb * (16 * XSTR) + row * XSTR + c) = xr0;
            if (has1) {
                int j = tid + 224;
                int r2 = j >> 4, c2 = (j & 15) << 2;
                *(float4*)(lds + OFF_X + pb * (16 * XSTR) + r2 * XSTR + c2) = xr1;
            }
        }

        // tanh + store h1_t into CURRENT buffer (no WAR: prev buffer untouched)
        #pragma unroll
        for (int r = 0; r < 8; ++r) {
            float v = TANHF(c0[r] + c1[r]);
            lds[OFF_H1 + cb * HBUF + (r + crow) * HSTR + wv * 16 + mi] = v;
        }
        __syncthreads();   // SINGLE barrier per step:
                           //  - h1[cur] visible for layer 2
                           //  - (one iter later) h2[cur] + staged x visible for next layer 1

        // --- layer 2: acc = h1[cur] @ W2x + h2[prev] @ W2h + b2 ---
        #pragma unroll
        for (int r = 0; r < 8; ++r) { c0[r] = bias2; c1[r] = 0.f; }
        {
            const float* hl = lds + OFF_H1 + cb * HBUF + mi * HSTR + kh;
            const float* wp = lds + OFF_W2X + wv * (KBH * 64) + lane * 2;
            #pragma unroll
            for (int k = 0; k < KBH; k += 2) {
                c0 = __builtin_amdgcn_wmma_f32_16x16x4_f32(
                        false, ld2(hl + 4 * k),       false, ld2(wp + 64 * k),
                        (short)0, c0, false, false);
                c1 = __builtin_amdgcn_wmma_f32_16x16x4_f32(
                        false, ld2(hl + 4 * (k + 1)), false, ld2(wp + 64 * (k + 1)),
                        (short)0, c1, false, false);
            }
            const float* h2l = lds + OFF_H2 + pb * HBUF + mi * HSTR + kh;
            wp = lds + OFF_W2H + wv * (KBH * 64) + lane * 2;
            #pragma unroll
            for (int k = 0; k < KBH; k += 2) {
                c0 = __builtin_amdgcn_wmma_f32_16x16x4_f32(
                        false, ld2(h2l + 4 * k),       false, ld2(wp + 64 * k),
                        (short)0, c0, false, false);
                c1 = __builtin_amdgcn_wmma_f32_16x16x4_f32(
                        false, ld2(h2l + 4 * (k + 1)), false, ld2(wp + 64 * (k + 1)),
                        (short)0, c1, false, false);
            }
        }

        // tanh + store h2_t into CURRENT buffer; no trailing barrier needed —
        // first reader is next iteration's layer 2, behind next iteration's barrier.
        #pragma unroll
        for (int r = 0; r < 8; ++r) {
            float v = TANHF(c0[r] + c1[r]);
            lds[OFF_H2 + cb * HBUF + (r + crow) * HSTR + wv * 16 + mi] = v;
        }
    }
    __syncthreads();

    // ---- epilogue: final states live in buffer (SEQT-1)&1 == 1 ----
    const int fb = (SEQT - 1) & 1;
    for (int i = tid; i < 16 * H; i += nthr) {
        int m = i / H, n = i - m * H;
        h1out[(size_t)(b0 + m) * H + n] = lds[OFF_H1 + fb * HBUF + m * HSTR + n];
        h2out[(size_t)(b0 + m) * H + n] = lds[OFF_H2 + fb * HBUF + m * HSTR + n];
    }
    if (tid < 16) {
        float s = bo[0];
        for (int n = 0; n < H; ++n) s += lds[OFF_H2 + fb * HBUF + tid * HSTR + n] * Wo[n];
        outp[b0 + tid] = s;
    }
}

extern "C" void kernel_launch(void* const* d_in, const int* in_sizes, int n_in,
                              void* d_out, int out_size, void* d_ws, size_t ws_size,
                              hipStream_t stream) {
    (void)in_sizes; (void)n_in; (void)out_size; (void)d_ws; (void)ws_size;
    const float* x   = (const float*)d_in[0];
    const float* W1x = (const float*)d_in[1];
    const float* W1h = (const float*)d_in[2];
    const float* b1  = (const float*)d_in[3];
    const float* W2x = (const float*)d_in[4];
    const float* W2h = (const float*)d_in[5];
    const float* b2  = (const float*)d_in[6];
    const float* Wo  = (const float*)d_in[7];
    const float* bo  = (const float*)d_in[8];

    float* outp  = (float*)d_out;            // [512,1]
    float* h1out = outp + BATCH;             // [512,100]
    float* h2out = h1out + BATCH * H;        // [512,100]

    const size_t shmem = (size_t)LDS_FLOATS * sizeof(float); // 217600 B < 320 KB
    (void)hipFuncSetAttribute((const void*)drnn_fused_kernel,
                              hipFuncAttributeMaxDynamicSharedMemorySize, (int)shmem);

    drnn_fused_kernel<<<BATCH / 16, 224, shmem, stream>>>(
        x, W1x, W1h, b1, W2x, W2h, b2, Wo, bo, outp, h1out, h2out);
}